// SimpleRNN_55078660604132
// MI455X (gfx1250) — compile-verified
//
#include <hip/hip_runtime.h>

// ---------------------------------------------------------------------------
// Plastic RNN for MI455X (gfx1250, wave32).
//
// Roofline: wx (4MB) + wh (8MB) are read+written every one of 64 steps
// -> ~1.5 GB of traffic -> ~66us floor at 23.3 TB/s HBM; the whole 12MB
// working set fits in the 192MB L2, so the real bound is L2 bandwidth.
// The recurrence is a batched GEMV with per-sample weights (1 FMA / 4B):
// f32 VALU FMA on float4 loads is optimal there. WMMA (f32 16x16x4) is used
// where a true GEMM exists: the output projection hs(512x512) @ W'(512x16),
// with the masked B matrix staged in LDS so the K-loop is load+ds+wmma only
// (no divergent exec-mask churn between WMMA issues).
// ---------------------------------------------------------------------------

typedef float v2f __attribute__((ext_vector_type(2)));
typedef float v8f __attribute__((ext_vector_type(8)));

static constexpr int kT = 64;
static constexpr int kB = 8;
static constexpr int kH = 512;
static constexpr int kI = 256;
static constexpr int kO = 8;
static constexpr int kG = 4;
static constexpr int kEH = 409;    // int(0.8*512): sign +1 below, -1 at/after
static constexpr int kEXN = 410;   // H - round(0.2*512): exist==1 below
static constexpr float kDT = 0.02f;
static constexpr float kAX = 0.2f;   // DT / 0.1
static constexpr float kAW = 0.02f;  // DT / 1.0

__device__ __forceinline__ float frelu(float v) { return v > 0.f ? v : 0.f; }

// ---------------------------------------------------------------------------
__global__ void zero_ws_kernel(float4* __restrict__ p, int n4) {
  int i = blockIdx.x * blockDim.x + threadIdx.x;
  if (i < n4) p[i] = make_float4(0.f, 0.f, 0.f, 0.f);
}

// ---------------------------------------------------------------------------
// One timestep. Grid: B * (H/16) blocks of 256 threads (8 waves).
// Block owns 16 rows (b, h0..h0+15): computes new_out for those rows and
// updates the matching wx/wh rows -> no cross-block hazard within a step.
// ---------------------------------------------------------------------------
__global__ __launch_bounds__(256) void rnn_step_kernel(
    const float* __restrict__ xt,        // (B,I)  x[t]
    const float* __restrict__ Rt,        // (B,)   Rs[t]
    const float* __restrict__ out_prev,  // (B,H)  hs[t-1] (or zeros)
    float* __restrict__ out_new,         // (B,H)  hs[t]
    float* __restrict__ state,           // (B,H)
    float* __restrict__ wx,              // (B,H,I)
    float* __restrict__ wh,              // (B,H,H)
    const float* __restrict__ Wx,        // (H,I)
    const float* __restrict__ Wh,        // (H,H)
    const float* __restrict__ bh,        // (H)
    const float* __restrict__ Wattn,     // (G,H)
    const float* __restrict__ battn,     // (G)
    const float* __restrict__ kappa)     // (6)
{
  const int b = blockIdx.x >> 5;           // 32 h-tiles per batch
  const int h0 = (blockIdx.x & 31) << 4;   // 16 rows per tile
  const int tid = threadIdx.x;

  __shared__ float s_out[kH];
  __shared__ float s_xin[kI];
  __shared__ float s_log[kG];

  // Stage previous output and raw input.
  s_out[tid]       = out_prev[b * kH + tid];
  s_out[tid + 256] = out_prev[b * kH + tid + 256];
  const float xraw = xt[b * kI + tid];
  __syncthreads();

  // Attention logits: one wave per gate g (4 waves), wave-reduce over H.
  if (tid < kG * 32) {
    const int g = tid >> 5, lane = tid & 31;
    float p = 0.f;
    for (int j = lane; j < kH; j += 32) {
      const float m = (j < kEXN) ? ((j < kEH) ? 1.f : -1.f) : 0.f;
      p += s_out[j] * frelu(Wattn[g * kH + j]) * m;
    }
    p += __shfl_xor(p, 1);
    p += __shfl_xor(p, 2);
    p += __shfl_xor(p, 4);
    p += __shfl_xor(p, 8);
    p += __shfl_xor(p, 16);
    if (lane == 0) s_log[g] = p + battn[g];
  }
  __syncthreads();

  // Softmax over G=4 gates; xin = relu(x) * af * G  (af = aw repeated GS=64).
  {
    const float l0 = s_log[0], l1 = s_log[1], l2 = s_log[2], l3 = s_log[3];
    const float mx = fmaxf(fmaxf(l0, l1), fmaxf(l2, l3));
    const float e0 = __expf(l0 - mx), e1 = __expf(l1 - mx);
    const float e2 = __expf(l2 - mx), e3 = __expf(l3 - mx);
    const float inv = 1.f / (e0 + e1 + e2 + e3);
    const int g = tid >> 6;  // GS = 64
    const float a = (g == 0 ? e0 : g == 1 ? e1 : g == 2 ? e2 : e3) * inv;
    s_xin[tid] = frelu(xraw) * a * (float)kG;
  }
  __syncthreads();

  // Row phase: 16 lanes per row, float4-vectorized streams of wx/wh rows.
  const int row = tid >> 4;
  const int l = tid & 15;
  const int h = h0 + row;

  float4* wxr = (float4*)(wx + (size_t)(b * kH + h) * kI);
  float4* whr = (float4*)(wh + (size_t)(b * kH + h) * kH);
  const float4* Wxr = (const float4*)(Wx + (size_t)h * kI);
  const float4* Whr = (const float4*)(Wh + (size_t)h * kH);
  const float4* xin4 = (const float4*)s_xin;
  const float4* out4 = (const float4*)s_out;

  float acc = 0.f;
#pragma unroll
  for (int it = 0; it < kI / 64; ++it) {  // 4 iters: 64 float4 per row
    const int q = l + 16 * it;
    const float4 w = Wxr[q], p = wxr[q], xv = xin4[q];
    acc += (frelu(w.x) + p.x) * xv.x + (frelu(w.y) + p.y) * xv.y +
           (frelu(w.z) + p.z) * xv.z + (frelu(w.w) + p.w) * xv.w;
  }
#pragma unroll
  for (int it = 0; it < kH / 64; ++it) {  // 8 iters: 128 float4 per row
    const int q = l + 16 * it;
    const int j = q << 2;
    const float4 w = Whr[q], p = whr[q], ov = out4[q];
    const float m0 = (j     == h) ? 0.f : ((j     < kEH) ? 1.f : -1.f);
    const float m1 = (j + 1 == h) ? 0.f : ((j + 1 < kEH) ? 1.f : -1.f);
    const float m2 = (j + 2 == h) ? 0.f : ((j + 2 < kEH) ? 1.f : -1.f);
    const float m3 = (j + 3 == h) ? 0.f : ((j + 3 < kEH) ? 1.f : -1.f);
    acc += (frelu(w.x) + p.x) * m0 * ov.x + (frelu(w.y) + p.y) * m1 * ov.y +
           (frelu(w.z) + p.z) * m2 * ov.z + (frelu(w.w) + p.w) * m3 * ov.w;
  }
  // Reduce across the 16 lanes of this row (xor <= 8 stays inside the group).
  acc += __shfl_xor(acc, 1);
  acc += __shfl_xor(acc, 2);
  acc += __shfl_xor(acc, 4);
  acc += __shfl_xor(acc, 8);

  const float total = acc + bh[h];
  const float ns = state[b * kH + h] * (1.f - kAX) + total * kAX;
  const float no = tanhf(fmaxf(ns, 0.f));
  if (l == 0) {
    state[b * kH + h] = ns;
    out_new[b * kH + h] = no;
  }

  // Plastic updates (second streamed pass; rows are hot in cache).
  const float R = Rt[b];
  const float dr = kDT * R;
  const float dec = 1.f - kAW;
  const float k0 = kappa[0], k1 = kappa[1], k2 = kappa[2];
  const float k3 = kappa[3], k4 = kappa[4], k5 = kappa[5];

#pragma unroll
  for (int it = 0; it < kI / 64; ++it) {
    const int q = l + 16 * it;
    const float4 w = Wxr[q], p = wxr[q], xv = xin4[q];
    float4 r;
    r.x = fmaxf(p.x * dec + dr * (k0 * xv.x + k1 * no + k2 * no * xv.x), -frelu(w.x));
    r.y = fmaxf(p.y * dec + dr * (k0 * xv.y + k1 * no + k2 * no * xv.y), -frelu(w.y));
    r.z = fmaxf(p.z * dec + dr * (k0 * xv.z + k1 * no + k2 * no * xv.z), -frelu(w.z));
    r.w = fmaxf(p.w * dec + dr * (k0 * xv.w + k1 * no + k2 * no * xv.w), -frelu(w.w));
    wxr[q] = r;
  }
#pragma unroll
  for (int it = 0; it < kH / 64; ++it) {
    const int q = l + 16 * it;
    const float4 w = Whr[q], p = whr[q], ov = out4[q];  // ov = OLD out
    float4 r;
    r.x = fmaxf(p.x * dec + dr * (k3 * ov.x + k4 * no + k5 * no * ov.x), -frelu(w.x));
    r.y = fmaxf(p.y * dec + dr * (k3 * ov.y + k4 * no + k5 * no * ov.y), -frelu(w.y));
    r.z = fmaxf(p.z * dec + dr * (k3 * ov.z + k4 * no + k5 * no * ov.z), -frelu(w.z));
    r.w = fmaxf(p.w * dec + dr * (k3 * ov.w + k4 * no + k5 * no * ov.w), -frelu(w.w));
    whr[q] = r;
  }
}

// ---------------------------------------------------------------------------
// Output projection: os(512 x 8) = hs(512 x 512) @ W'(512 x 16padded)
// via V_WMMA_F32_16X16X4_F32, one wave per 16-row tile, K-loop of 128 WMMAs.
// The masked/relu'd/zero-padded B matrix (512 x 16 = 32KB) is staged in LDS
// with a uniform fill so the K-loop has no divergence around the WMMAs.
// A layout (f32 16x4): lanes 0-15 hold K=0/1 in v[0]/v[1], lanes 16-31 K=2/3.
// B layout (f32 4x16): N = lane%16, same K striping across the half-waves.
// C/D: v[r] -> M = m0 + r (+8 for lanes 16-31), N = lane%16.
// ---------------------------------------------------------------------------
__global__ __launch_bounds__(32) void out_proj_kernel(
    const float* __restrict__ hs,   // (T*B, H) row-major
    const float* __restrict__ Who,  // (O, H)
    float* __restrict__ os)         // (T*B, O)
{
  __shared__ float s_w[kH][16];     // s_w[k][n] = relu(Who[n][k]) * exist(k)

  const int L = threadIdx.x;

  // Uniform, non-divergent fill: 8 real columns + 8 zero-pad columns.
  for (int i = L; i < kO * kH; i += 32) {   // 4096 elems, 128 per lane
    const int o = i >> 9;                   // i / kH
    const int k = i & (kH - 1);             // i % kH
    s_w[k][o]      = (k < kEXN) ? frelu(Who[i]) : 0.f;
    s_w[k][o + 8]  = 0.f;
  }
  __syncthreads();

  const int m0 = blockIdx.x << 4;
  const int half = L >> 4;   // 0: K+0/K+1, 1: K+2/K+3
  const int n = L & 15;      // B column / A row-within-tile

  v8f c = {};
  const float* arow = hs + (size_t)(m0 + n) * kH;
#pragma unroll 4
  for (int kk = 0; kk < kH / 4; ++kk) {
    const int k0 = (kk << 2) + (half << 1);
    v2f a, bm;
    a.x = arow[k0];
    a.y = arow[k0 + 1];
    bm.x = s_w[k0][n];
    bm.y = s_w[k0 + 1][n];
    c = __builtin_amdgcn_wmma_f32_16x16x4_f32(false, a, false, bm,
                                              (short)0, c, false, false);
  }
#pragma unroll
  for (int r = 0; r < 8; ++r) {
    const int m = m0 + r + (half << 3);
    if (n < kO) os[m * kO + n] = c[r];
  }
}

// ---------------------------------------------------------------------------
extern "C" void kernel_launch(void* const* d_in, const int* in_sizes, int n_in,
                              void* d_out, int out_size, void* d_ws,
                              size_t ws_size, hipStream_t stream) {
  const float* x     = (const float*)d_in[0];  // (T,B,I)
  const float* Rs    = (const float*)d_in[1];  // (T,B,1)
  const float* Wx    = (const float*)d_in[2];  // (H,I)
  const float* Wh    = (const float*)d_in[3];  // (H,H)
  const float* bhb   = (const float*)d_in[4];  // (H)
  const float* Who   = (const float*)d_in[5];  // (O,H)
  const float* Wattn = (const float*)d_in[6];  // (G,H)
  const float* battn = (const float*)d_in[7];  // (G)
  const float* kappa = (const float*)d_in[8];  // (6)

  float* os = (float*)d_out;            // (T,B,O)
  float* hs = os + kT * kB * kO;        // (T,B,H); hs[t] is also out-chain

  float* ws    = (float*)d_ws;          // ~12.03 MB used
  float* state = ws;                    // (B,H)
  float* zeros = ws + kB * kH;          // (B,H) stays zero: out at t=0
  float* wx    = ws + 2 * kB * kH;      // (B,H,I)
  float* wh    = wx + (size_t)kB * kH * kI;  // (B,H,H)

  const size_t zfloats =
      (size_t)2 * kB * kH + (size_t)kB * kH * kI + (size_t)kB * kH * kH;
  const int n4 = (int)(zfloats / 4);
  zero_ws_kernel<<<(n4 + 255) / 256, 256, 0, stream>>>((float4*)d_ws, n4);

  for (int t = 0; t < kT; ++t) {
    const float* outp = (t == 0) ? zeros : (hs + (size_t)(t - 1) * kB * kH);
    rnn_step_kernel<<<kB * (kH / 16), 256, 0, stream>>>(
        x + (size_t)t * kB * kI, Rs + (size_t)t * kB, outp,
        hs + (size_t)t * kB * kH, state, wx, wh, Wx, Wh, bhb, Wattn, battn,
        kappa);
  }
  out_proj_kernel<<<(kT * kB) / 16, 32, 0, stream>>>(hs, Who, os);
}